// GCN_76407468195987
// MI455X (gfx1250) — compile-verified
//
#include <hip/hip_runtime.h>

// GCN: 3x (GEMM + gcn_norm scatter-add) + global add pool + linear head
// Sizes (from reference): N=100000, E=640000, D=128, L=3, B=128, OUT=10

#define D 128
#define LDSP 132  // col-major LDS pitch; 132 % 64 == 4 -> conflict-free WMMA B reads

typedef __attribute__((ext_vector_type(2))) float v2f;
typedef __attribute__((ext_vector_type(8))) float v8f;

// ---------------- degree / normalization ----------------
__global__ void k_init_deg(float* __restrict__ dinv, int n) {
  int i = blockIdx.x * blockDim.x + threadIdx.x;
  if (i < n) dinv[i] = 1.0f;  // self-loop contributes 1 to every node's degree
}

__global__ void k_count_deg(const int* __restrict__ dst, float* __restrict__ dinv, int e) {
  int i = blockIdx.x * blockDim.x + threadIdx.x;
  if (i < e) atomicAdd(&dinv[dst[i]], 1.0f);
}

__global__ void k_finish_deg(float* __restrict__ dinv, int n) {
  int i = blockIdx.x * blockDim.x + threadIdx.x;
  if (i < n) dinv[i] = rsqrtf(dinv[i]);  // deg >= 1 always (self-loops)
}

// ---------------- dense transform: HW = H @ W, fp32 WMMA ----------------
// Block = 256 threads = 8 waves; wave w computes rows [blk*128 + w*16, +16) x all 128 cols.
// W staged in LDS column-major: sW[c*LDSP + k] = W[k*D + c].
__global__ void k_gemm(const float* __restrict__ H, const float* __restrict__ W,
                       float* __restrict__ HW, int nrows) {
  __shared__ float sW[D * LDSP];  // 67,584 B
  for (int idx = threadIdx.x; idx < D * D; idx += blockDim.x) {
    int k = idx >> 7, c = idx & (D - 1);
    sW[c * LDSP + k] = W[idx];
  }
  __syncthreads();

  const int wave = threadIdx.x >> 5;
  const int lane = threadIdx.x & 31;
  const int row0 = (blockIdx.x * 8 + wave) * 16;
  if (row0 >= nrows) return;  // wave-uniform: N is a multiple of 16, EXEC stays all-1s

  const int m     = lane & 15;          // A: row / B,C,D: column-within-tile
  const int khalf = (lane >> 4) << 1;   // lanes 16..31 hold K=+2,+3
  const int mrow  = (lane >> 4) << 3;   // C/D: lanes 16..31 hold M=r+8

  v8f acc[8] = {};
  const float* arow = H + (size_t)(row0 + m) * D + khalf;

  for (int k0 = 0; k0 < D; k0 += 4) {
    v2f a = *(const v2f*)(arow + k0);  // {A[m][k0+khalf], A[m][k0+khalf+1]}
    const float* wb = sW + k0 + khalf; // + col*LDSP below
#pragma unroll
    for (int t = 0; t < 8; ++t) {
      // B fragment: {W[k0+khalf][col], W[k0+khalf+1][col]} -> one ds_load_b64
      v2f b = *(const v2f*)(wb + (size_t)(t * 16 + m) * LDSP);
      acc[t] = __builtin_amdgcn_wmma_f32_16x16x4_f32(
          false, a, false, b, (short)0, acc[t], false, false);
    }
  }

  float* out = HW + (size_t)(row0 + mrow) * D + m;
#pragma unroll
  for (int t = 0; t < 8; ++t)
#pragma unroll
    for (int r = 0; r < 8; ++r)
      out[(size_t)r * D + t * 16] = acc[t][r];
}

// ---------------- aggregation ----------------
// Self-loop message initializes agg (also zero-fills workspace deterministically).
__global__ void k_self_agg(const float* __restrict__ hw, const float* __restrict__ dinv,
                           float* __restrict__ agg, int total) {
  int i = blockIdx.x * blockDim.x + threadIdx.x;
  if (i < total) {
    float dv = dinv[i >> 7];
    agg[i] = hw[i] * dv * dv;
  }
}

// One wave per edge: float4 gather from hw[src], scale, 4 atomic f32 adds into agg[dst].
__global__ void k_edge_agg(const int* __restrict__ src, const int* __restrict__ dst,
                           const float* __restrict__ dinv, const float* __restrict__ hw,
                           float* __restrict__ agg, int e) {
  int edge = blockIdx.x * (blockDim.x >> 5) + (threadIdx.x >> 5);
  if (edge >= e) return;
  int lane = threadIdx.x & 31;
  int s = src[edge], d = dst[edge];
  float nrm = dinv[s] * dinv[d];
  float4 v = *(const float4*)(hw + (size_t)s * D + (lane << 2));
  float* o = agg + (size_t)d * D + (lane << 2);
  atomicAdd(o + 0, v.x * nrm);
  atomicAdd(o + 1, v.y * nrm);
  atomicAdd(o + 2, v.z * nrm);
  atomicAdd(o + 3, v.w * nrm);
}

__global__ void k_bias_relu(float* __restrict__ h, const float* __restrict__ b, int total) {
  int i = blockIdx.x * blockDim.x + threadIdx.x;
  if (i < total) {
    float v = h[i] + b[i & (D - 1)];
    h[i] = v > 0.0f ? v : 0.0f;
  }
}

// ---------------- pooling + head ----------------
__global__ void k_zero(float* __restrict__ p, int n) {
  int i = blockIdx.x * blockDim.x + threadIdx.x;
  if (i < n) p[i] = 0.0f;
}

__global__ void k_pool(const float* __restrict__ h, const int* __restrict__ batch,
                       float* __restrict__ pooled, int total) {
  int i = blockIdx.x * blockDim.x + threadIdx.x;
  if (i < total) atomicAdd(&pooled[batch[i >> 7] * D + (i & (D - 1))], h[i]);
}

__global__ void k_head(const float* __restrict__ pooled, const float* __restrict__ Wm,
                       const float* __restrict__ bm, float* __restrict__ out,
                       int nb, int nout) {
  int i = blockIdx.x * blockDim.x + threadIdx.x;
  if (i >= nb * nout) return;
  int b = i / nout, o = i - b * nout;
  float acc = bm[o];
  for (int d = 0; d < D; ++d) acc += pooled[b * D + d] * Wm[d * nout + o];
  out[i] = acc;
}

// ---------------- launch ----------------
extern "C" void kernel_launch(void* const* d_in, const int* in_sizes, int n_in,
                              void* d_out, int out_size, void* d_ws, size_t ws_size,
                              hipStream_t stream) {
  const float* x     = (const float*)d_in[0];
  const int*   ei    = (const int*)d_in[1];  // [2, E]: row 0 = src, row 1 = dst
  const int*   batch = (const int*)d_in[2];
  const float* Ws    = (const float*)d_in[3];  // [L, D, D]
  const float* bs    = (const float*)d_in[4];  // [L, D]
  const float* Wm    = (const float*)d_in[5];  // [D, OUT]
  const float* bm    = (const float*)d_in[6];  // [OUT]
  float* out = (float*)d_out;

  const int N   = in_sizes[0] / D;
  const int E   = in_sizes[1] / 2;
  const int L   = in_sizes[3] / (D * D);
  const int OUT = in_sizes[6];
  const int B   = out_size / OUT;
  const int ND  = N * D;

  // workspace layout: dinv[N] | bufA[N*D] | bufB[N*D] | hw[N*D] | pooled[B*D]
  float* dinv   = (float*)d_ws;
  float* bufA   = dinv + N;
  float* bufB   = bufA + (size_t)ND;
  float* hw     = bufB + (size_t)ND;
  float* pooled = hw + (size_t)ND;

  const int T = 256;

  // GCN normalization: deg (with self-loops) -> dinv = rsqrt(deg)
  k_init_deg<<<(N + T - 1) / T, T, 0, stream>>>(dinv, N);
  k_count_deg<<<(E + T - 1) / T, T, 0, stream>>>(ei + E, dinv, E);
  k_finish_deg<<<(N + T - 1) / T, T, 0, stream>>>(dinv, N);

  const float* h_in = x;
  float* bufs[2] = {bufA, bufB};
  const float* cur = x;
  for (int l = 0; l < L; ++l) {
    float* agg = bufs[l & 1];
    k_gemm<<<(N + 127) / 128, 256, 0, stream>>>(h_in, Ws + (size_t)l * D * D, hw, N);
    k_self_agg<<<(ND + T - 1) / T, T, 0, stream>>>(hw, dinv, agg, ND);
    k_edge_agg<<<(E + 7) / 8, 256, 0, stream>>>(ei, ei + E, dinv, hw, agg, E);
    k_bias_relu<<<(ND + T - 1) / T, T, 0, stream>>>(agg, bs + (size_t)l * D, ND);
    h_in = agg;
    cur  = agg;
  }

  k_zero<<<(B * D + T - 1) / T, T, 0, stream>>>(pooled, B * D);
  k_pool<<<(ND + T - 1) / T, T, 0, stream>>>(cur, batch, pooled, ND);
  k_head<<<(B * OUT + T - 1) / T, T, 0, stream>>>(pooled, Wm, bm, out, B, OUT);
}